// GATConvBlock_3032246911688
// MI455X (gfx1250) — compile-verified
//
#include <hip/hip_runtime.h>
#include <math.h>

// ---------------------------------------------------------------------------
// GATConv (H=4 heads, C=32) + bias + BatchNorm1d (batch stats) + LeakyReLU
// for N=100000 nodes, E=1.6M edges, F_IN=F_OUT=128, on gfx1250 (CDNA5).
//
// Pipeline:
//   0) init        : zero out / denom / stats, set segment-max to enc(-inf)
//   1) gemm_wmma   : xt = x @ W^T via V_WMMA_F32_16X16X4_F32 (fp32 exact)
//   2) logits      : a_src[n,h], a_dst[n,h] per-head dot products
//   3) edge_max    : e = leaky(a_src[src]+a_dst[dst]); atomicMax per dst
//   4) edge_exp    : p = exp(e - m[dst]); atomicAdd denom[dst]
//   5) aggregate   : out[dst] += (p/denom[dst]) * xt[src]  (32 lanes/edge)
//   6) stats       : per-channel sum / sumsq (block partials + atomics)
//   7) bn_act      : (v-mean)*rsqrt(var+eps)*gamma+beta, leaky_relu
// ---------------------------------------------------------------------------

#define N_NODES 100000
#define N_EDGES 1600000
#define F_IN    128
#define F_OUT   128
#define NHEAD   4
#define CHAN    32
#define NEG_SLOPE 0.2f
#define BN_EPS  1e-5f

typedef __attribute__((ext_vector_type(2))) float v2f;
typedef __attribute__((ext_vector_type(8))) float v8f;

__device__ __forceinline__ float leaky(float x) { return x > 0.f ? x : NEG_SLOPE * x; }

// Order-preserving float -> uint encoding so unsigned atomicMax == float max.
__device__ __forceinline__ unsigned fenc(float f) {
  unsigned u = __float_as_uint(f);
  return (u & 0x80000000u) ? ~u : (u | 0x80000000u);
}
__device__ __forceinline__ float fdec(unsigned u) {
  return __uint_as_float((u & 0x80000000u) ? (u & 0x7FFFFFFFu) : ~u);
}
#define ENC_NEG_INF 0x007FFFFFu  // fenc(-inf)

// ---------------- 1) xt = x @ W^T, fp32 WMMA 16x16x4 ----------------
// One wave computes a 16x16 tile of xt. Block = 256 threads = 8 waves,
// wave w covers output columns [16w,16w+16); blockIdx.x covers rows.
// A frag (16x4 f32, 2 VGPR): lane L -> M=L%16, K=(L/16)*2+j
// B frag (4x16 f32, 2 VGPR): lane L -> N=L%16, K=(L/16)*2+j
// D (16x16 f32, 8 VGPR):     lane L -> N=L%16, M=j+8*(L/16)
__global__ __launch_bounds__(256) void k_gemm_wmma(
    const float* __restrict__ x, const float* __restrict__ W,
    float* __restrict__ xt) {
  const int wave = threadIdx.x >> 5;
  const int lane = threadIdx.x & 31;
  const int rowBase = blockIdx.x * 16;
  const int colBase = wave * 16;
  const int m  = lane & 15;
  const int kb = (lane >> 4) << 1;

  const float* __restrict__ xrow = x + (size_t)(rowBase + m) * F_IN;
  const float* __restrict__ wrow = W + (size_t)(colBase + m) * F_IN;  // W[o][k]

  v8f acc = {};
#pragma unroll 4
  for (int k = 0; k < F_IN; k += 4) {
    v2f a = *(const v2f*)(xrow + k + kb);   // x[rowBase+m][k+kb .. +1]
    v2f b = *(const v2f*)(wrow + k + kb);   // W[colBase+m][k+kb .. +1] = (W^T)[k+kb][col]
    acc = __builtin_amdgcn_wmma_f32_16x16x4_f32(
        /*neg_a=*/false, a, /*neg_b=*/false, b,
        /*c_mod=*/(short)0, acc, /*reuse_a=*/false, /*reuse_b=*/false);
  }

  const int n     = colBase + m;
  const int mrow0 = rowBase + ((lane >> 4) << 3);
#pragma unroll
  for (int j = 0; j < 8; ++j)
    xt[(size_t)(mrow0 + j) * F_OUT + n] = acc[j];
}

// ---------------- 0) init ----------------
__global__ void k_init(float* __restrict__ out, unsigned* __restrict__ menc,
                       float* __restrict__ denom, float* __restrict__ stats) {
  const int t = blockIdx.x * blockDim.x + threadIdx.x;
  if (t < N_NODES * F_OUT) out[t] = 0.f;
  if (t < N_NODES * NHEAD) { menc[t] = ENC_NEG_INF; denom[t] = 0.f; }
  if (t < 2 * F_OUT) stats[t] = 0.f;
}

// ---------------- 2) per-node attention logits ----------------
__global__ void k_logits(const float* __restrict__ xt,
                         const float* __restrict__ att_src,
                         const float* __restrict__ att_dst,
                         float* __restrict__ a_src, float* __restrict__ a_dst) {
  const int t = blockIdx.x * blockDim.x + threadIdx.x;
  if (t >= N_NODES * NHEAD) return;
  const int n = t >> 2, h = t & 3;
  const float* __restrict__ xp = xt + (size_t)n * F_OUT + h * CHAN;
  const float* __restrict__ as = att_src + h * CHAN;
  const float* __restrict__ ad = att_dst + h * CHAN;
  float s = 0.f, d = 0.f;
#pragma unroll
  for (int c = 0; c < CHAN; ++c) { const float v = xp[c]; s += v * as[c]; d += v * ad[c]; }
  a_src[t] = s;
  a_dst[t] = d;
}

// ---------------- 3) per-edge score + segment max ----------------
__global__ void k_edge_max(const int* __restrict__ ei,
                           const float* __restrict__ a_src,
                           const float* __restrict__ a_dst,
                           float* __restrict__ ebuf, unsigned* __restrict__ menc) {
  const int e = blockIdx.x * blockDim.x + threadIdx.x;
  if (e >= N_EDGES) return;
  const int s = ei[e], d = ei[N_EDGES + e];
  const float4 vs = *(const float4*)(a_src + (size_t)s * NHEAD);
  const float4 vd = *(const float4*)(a_dst + (size_t)d * NHEAD);
  float ev[4] = { leaky(vs.x + vd.x), leaky(vs.y + vd.y),
                  leaky(vs.z + vd.z), leaky(vs.w + vd.w) };
#pragma unroll
  for (int h = 0; h < NHEAD; ++h) {
    ebuf[(size_t)e * NHEAD + h] = ev[h];
    atomicMax(&menc[(size_t)d * NHEAD + h], fenc(ev[h]));
  }
}

// ---------------- 4) exp + segment sum ----------------
__global__ void k_edge_exp(const int* __restrict__ ei, float* __restrict__ ebuf,
                           const unsigned* __restrict__ menc,
                           float* __restrict__ denom) {
  const int e = blockIdx.x * blockDim.x + threadIdx.x;
  if (e >= N_EDGES) return;
  const int d = ei[N_EDGES + e];
#pragma unroll
  for (int h = 0; h < NHEAD; ++h) {
    const float mx = fdec(menc[(size_t)d * NHEAD + h]);
    const float p  = expf(ebuf[(size_t)e * NHEAD + h] - mx);
    ebuf[(size_t)e * NHEAD + h] = p;
    atomicAdd(&denom[(size_t)d * NHEAD + h], p);
  }
}

// ---------------- 5) weighted scatter aggregation ----------------
// 32 lanes per edge; each lane handles 4 contiguous channels (float4 gather +
// 4 scalar f32 atomic adds). This pass is the HBM roofline bottleneck.
__global__ __launch_bounds__(256) void k_aggregate(
    const int* __restrict__ ei, const float* __restrict__ ebuf,
    const float* __restrict__ denom, const float* __restrict__ xt,
    float* __restrict__ out) {
  const int t = blockIdx.x * blockDim.x + threadIdx.x;
  const int e = t >> 5;
  const int lane = t & 31;
  if (e >= N_EDGES) return;
  const int s = ei[e], d = ei[N_EDGES + e];
  const int f0 = lane << 2;        // channel base, 0..124
  const int h  = lane >> 3;        // head for these 4 channels
  const float p  = ebuf[(size_t)e * NHEAD + h];
  const float dn = denom[(size_t)d * NHEAD + h];
  const float alpha = p / dn;
  const float4 v = *(const float4*)(xt + (size_t)s * F_OUT + f0);
  float* __restrict__ o = out + (size_t)d * F_OUT + f0;
  atomicAdd(o + 0, alpha * v.x);
  atomicAdd(o + 1, alpha * v.y);
  atomicAdd(o + 2, alpha * v.z);
  atomicAdd(o + 3, alpha * v.w);
}

// ---------------- 6) batchnorm statistics ----------------
__global__ __launch_bounds__(128) void k_stats(const float* __restrict__ out,
                                               const float* __restrict__ bias,
                                               float* __restrict__ stats) {
  const int c = threadIdx.x;         // one thread per channel
  const float b = bias[c];
  float s = 0.f, s2 = 0.f;
  for (int r = blockIdx.x; r < N_NODES; r += gridDim.x) {
    const float v = out[(size_t)r * F_OUT + c] + b;
    s += v; s2 += v * v;
  }
  atomicAdd(&stats[c], s);
  atomicAdd(&stats[F_OUT + c], s2);
}

// ---------------- 7) normalize + activation ----------------
__global__ void k_bn_act(float* __restrict__ out, const float* __restrict__ bias,
                         const float* __restrict__ gamma,
                         const float* __restrict__ beta,
                         const float* __restrict__ stats) {
  const int t = blockIdx.x * blockDim.x + threadIdx.x;
  if (t >= N_NODES * F_OUT) return;
  const int c = t & (F_OUT - 1);
  const float inv_n = 1.f / (float)N_NODES;
  const float mean = stats[c] * inv_n;
  const float var  = stats[F_OUT + c] * inv_n - mean * mean;
  const float v = out[t] + bias[c];
  const float y = (v - mean) * rsqrtf(var + BN_EPS) * gamma[c] + beta[c];
  out[t] = leaky(y);
}

// ---------------------------------------------------------------------------
extern "C" void kernel_launch(void* const* d_in, const int* in_sizes, int n_in,
                              void* d_out, int out_size, void* d_ws, size_t ws_size,
                              hipStream_t stream) {
  const float* x       = (const float*)d_in[0];  // [N, 128]
  const float* W       = (const float*)d_in[1];  // [128, 128]
  const float* att_src = (const float*)d_in[2];  // [4, 32]
  const float* att_dst = (const float*)d_in[3];  // [4, 32]
  const float* bias    = (const float*)d_in[4];  // [128]
  const float* gamma   = (const float*)d_in[5];  // [128]
  const float* beta    = (const float*)d_in[6];  // [128]
  const int*   ei      = (const int*)d_in[7];    // [2, E]
  float* out = (float*)d_out;                    // [N, 128]

  // Workspace carve-up (~83 MB)
  float*    xt    = (float*)d_ws;                          // N*128
  float*    ebuf  = xt    + (size_t)N_NODES * F_OUT;       // E*4
  float*    a_src = ebuf  + (size_t)N_EDGES * NHEAD;       // N*4
  float*    a_dst = a_src + (size_t)N_NODES * NHEAD;       // N*4
  unsigned* menc  = (unsigned*)(a_dst + (size_t)N_NODES * NHEAD); // N*4
  float*    denom = (float*)(menc + (size_t)N_NODES * NHEAD);     // N*4
  float*    stats = denom + (size_t)N_NODES * NHEAD;              // 256

  k_init<<<(N_NODES * F_OUT + 255) / 256, 256, 0, stream>>>(out, menc, denom, stats);
  k_gemm_wmma<<<N_NODES / 16, 256, 0, stream>>>(x, W, xt);                  // 6250 blocks
  k_logits<<<(N_NODES * NHEAD + 255) / 256, 256, 0, stream>>>(xt, att_src, att_dst, a_src, a_dst);
  k_edge_max<<<N_EDGES / 256, 256, 0, stream>>>(ei, a_src, a_dst, ebuf, menc);
  k_edge_exp<<<N_EDGES / 256, 256, 0, stream>>>(ei, ebuf, menc, denom);
  k_aggregate<<<(N_EDGES * 32) / 256, 256, 0, stream>>>(ei, ebuf, denom, xt, out);
  k_stats<<<512, 128, 0, stream>>>(out, bias, stats);
  k_bn_act<<<(N_NODES * F_OUT + 255) / 256, 256, 0, stream>>>(out, bias, gamma, beta, stats);
}